// Detector_80659485819530
// MI455X (gfx1250) — compile-verified
//
#include <hip/hip_runtime.h>
#include <math.h>

// ---------------- problem constants (match reference) ----------------
#define B_        8
#define NBOX      300
#define NCLS      81          // includes background class 0
#define NC        80          // foreground classes
#define MAXDET    30
#define K_        32          // whole + 30 + whole
#define OUTSZ     256
#define SCALE_F   600.0f
#define NMS_THF   0.3f
#define SCORE_THF 0.9f
#define ROWS_PB   16          // output rows per crop block (K4 pipeline depth)

// ---------------- workspace layout (floats unless noted) -------------
#define PRED_ELEMS ((size_t)B_*NBOX*NCLS*4)       // decoded boxes: 777600 f32
#define CNT_ELEMS  (B_*NC)                        // per-(b,c) kept count: 640 i32
#define KEPT_ELEMS ((size_t)B_*NC*MAXDET*4)       // compacted kept boxes
#define FBX_ELEMS  (B_*K_*4)                      // assembled fboxes
#define CROP_ELEMS ((size_t)B_*K_*3*OUTSZ*OUTSZ)  // output crops: 50,331,648 f32

// =====================================================================
// K1: bbox_transform_inv + clip.  One thread per (b,n,cls).
// =====================================================================
__global__ void k1_transform(const float* __restrict__ boxes,
                             const float* __restrict__ bbox_pred,
                             float* __restrict__ pred) {
  int tid = blockIdx.x * blockDim.x + threadIdx.x;
  if (tid >= B_ * NBOX * NCLS) return;
  int k  = tid % NCLS;
  int bn = tid / NCLS;

  const float* bx = boxes + (size_t)bn * 4;
  float x1 = bx[0], y1 = bx[1], x2 = bx[2], y2 = bx[3];
  float w  = x2 - x1 + 1.0f;
  float h  = y2 - y1 + 1.0f;
  float cx = x1 + 0.5f * w;
  float cy = y1 + 0.5f * h;

  const float* dp = bbox_pred + (size_t)bn * (4 * NCLS) + 4 * k;
  float dx = dp[0] * 0.1f, dy = dp[1] * 0.1f;
  float dw = dp[2] * 0.2f, dh = dp[3] * 0.2f;

  float pcx = dx * w + cx;
  float pcy = dy * h + cy;
  float pw  = expf(dw) * w;
  float ph  = expf(dh) * h;

  float lo = 0.0f, hi = SCALE_F - 1.0f;
  float4 o;
  o.x = fminf(fmaxf(pcx - 0.5f * pw, lo), hi);
  o.y = fminf(fmaxf(pcy - 0.5f * ph, lo), hi);
  o.z = fminf(fmaxf(pcx + 0.5f * pw, lo), hi);
  o.w = fminf(fmaxf(pcy + 0.5f * ph, lo), hi);
  ((float4*)pred)[tid] = o;
}

// =====================================================================
// K2: per (b, class) — bitonic sort 300(+pad) scores desc (stable),
// greedy NMS on valid prefix, compact kept boxes.
// blockIdx.x = b*80 + c, 256 threads (8 wave32).
// =====================================================================
__global__ void k2_sort_nms(const float* __restrict__ scores,
                            const float* __restrict__ pred,
                            float* __restrict__ kept,
                            int* __restrict__ cnt) {
  const int bc  = blockIdx.x;
  const int b   = bc / NC;
  const int c   = bc % NC;
  const int k   = c + 1;            // skip background
  const int tid = threadIdx.x;

  __shared__ float s_sc[512];
  __shared__ int   s_idx[512];
  __shared__ float s_box[NBOX * 4];
  __shared__ int   s_supp[NBOX];
  __shared__ int   s_nv;

  for (int e = tid; e < 512; e += 256) {
    if (e < NBOX) {
      s_sc[e]  = scores[((size_t)(b * NBOX + e)) * NCLS + k];
      s_idx[e] = e;
    } else {
      s_sc[e]  = -1e30f;
      s_idx[e] = e;
    }
  }

  // bitonic sort: final order = score desc, index asc on ties (stable argsort(-s))
  for (int size = 2; size <= 512; size <<= 1) {
    for (int stride = size >> 1; stride > 0; stride >>= 1) {
      __syncthreads();
      for (int e = tid; e < 512; e += 256) {
        int p = e ^ stride;
        if (p > e) {
          bool desc  = ((e & size) == 0);
          float sa = s_sc[e], sb = s_sc[p];
          int   ia = s_idx[e], ib = s_idx[p];
          bool aFirst = (sa > sb) || (sa == sb && ia < ib);
          if (desc ? !aFirst : aFirst) {
            s_sc[e] = sb; s_sc[p] = sa;
            s_idx[e] = ib; s_idx[p] = ia;
          }
        }
      }
    }
  }
  __syncthreads();

  if (tid == 0) s_nv = 0;
  __syncthreads();

  // gather sorted boxes; count valid prefix (scores sorted desc -> contiguous)
  for (int e = tid; e < NBOX; e += 256) {
    int n = s_idx[e];
    const float* pb = pred + (((size_t)(b * NBOX + n)) * NCLS + k) * 4;
    s_box[e * 4 + 0] = pb[0];
    s_box[e * 4 + 1] = pb[1];
    s_box[e * 4 + 2] = pb[2];
    s_box[e * 4 + 3] = pb[3];
    s_supp[e] = 0;
    if (s_sc[e] > SCORE_THF) atomicAdd(&s_nv, 1);
  }
  __syncthreads();
  const int nv = s_nv;

  // greedy NMS: serial over candidates, parallel suppression
  for (int i = 0; i < nv; ++i) {
    __syncthreads();
    if (s_supp[i]) continue;               // uniform read -> no divergent sync
    float ax1 = s_box[i * 4 + 0], ay1 = s_box[i * 4 + 1];
    float ax2 = s_box[i * 4 + 2], ay2 = s_box[i * 4 + 3];
    float aarea = (ax2 - ax1 + 1.0f) * (ay2 - ay1 + 1.0f);
    for (int e = i + 1 + tid; e < nv; e += 256) {
      if (s_supp[e]) continue;             // benign race (monotone 0->1)
      float bx1 = s_box[e * 4 + 0], by1 = s_box[e * 4 + 1];
      float bx2 = s_box[e * 4 + 2], by2 = s_box[e * 4 + 3];
      float iw = fmaxf(fminf(ax2, bx2) - fmaxf(ax1, bx1) + 1.0f, 0.0f);
      float ih = fmaxf(fminf(ay2, by2) - fmaxf(ay1, by1) + 1.0f, 0.0f);
      float inter = iw * ih;
      float barea = (bx2 - bx1 + 1.0f) * (by2 - by1 + 1.0f);
      float iou = inter / (aarea + barea - inter);
      if (iou > NMS_THF) s_supp[e] = 1;
    }
  }
  __syncthreads();

  if (tid == 0) {
    int cc = 0;
    float* kp = kept + (size_t)bc * MAXDET * 4;
    for (int i = 0; i < nv; ++i) {
      if (!s_supp[i]) {
        if (cc < MAXDET) {
          kp[cc * 4 + 0] = s_box[i * 4 + 0];
          kp[cc * 4 + 1] = s_box[i * 4 + 1];
          kp[cc * 4 + 2] = s_box[i * 4 + 2];
          kp[cc * 4 + 3] = s_box[i * 4 + 3];
        }
        cc++;                              // full count (cumsum semantics)
      }
    }
    cnt[bc] = cc;
  }
}

// =====================================================================
// K3: per batch — class-order prefix sum, scatter into 32-slot fboxes,
// emit bbox_num and fboxes outputs.  blockIdx.x = b, 128 threads.
// =====================================================================
__global__ void k3_assemble(const float* __restrict__ kept,
                            const int* __restrict__ cnt,
                            float* __restrict__ fbx,
                            int* __restrict__ tot,
                            float* __restrict__ d_out) {
  const int b = blockIdx.x;
  const int tid = threadIdx.x;
  __shared__ int s_pre[NC];
  __shared__ int s_tot;

  if (tid == 0) {
    int run = 0;
    for (int c = 0; c < NC; ++c) { s_pre[c] = run; run += cnt[b * NC + c]; }
    s_tot = run < MAXDET ? run : MAXDET;
  }
  __syncthreads();

  float* fb = fbx + (size_t)b * K_ * 4;
  for (int s = tid; s < K_; s += 128) {
    if (s == 0 || s == K_ - 1) {
      fb[s * 4 + 0] = 0.0f; fb[s * 4 + 1] = 0.0f;
      fb[s * 4 + 2] = SCALE_F; fb[s * 4 + 3] = SCALE_F;
    } else {
      fb[s * 4 + 0] = 0.0f; fb[s * 4 + 1] = 0.0f;
      fb[s * 4 + 2] = 0.0f; fb[s * 4 + 3] = 0.0f;
    }
  }
  __syncthreads();

  for (int c = tid; c < NC; c += 128) {
    int m = cnt[b * NC + c];
    if (m > MAXDET) m = MAXDET;
    const float* kp = kept + (size_t)(b * NC + c) * MAXDET * 4;
    for (int r = 0; r < m; ++r) {
      int slot = s_pre[c] + r;
      if (slot < MAXDET) {
        float* d = fb + (size_t)(1 + slot) * 4;
        d[0] = kp[r * 4 + 0]; d[1] = kp[r * 4 + 1];
        d[2] = kp[r * 4 + 2]; d[3] = kp[r * 4 + 3];
      }
    }
  }
  __syncthreads();

  if (tid == 0) {
    tot[b] = s_tot;
    d_out[CROP_ELEMS + b] = (float)(s_tot + 2);      // bbox_num
  }
  for (int e = tid; e < K_ * 4; e += 128)            // fboxes output
    d_out[CROP_ELEMS + B_ + (size_t)b * K_ * 4 + e] = fb[e];
}

// =====================================================================
// K4: crop gather with double-buffered CDNA5 async global->LDS pipeline.
// blockIdx = (row-chunk, crop).  256 threads = output cols.
// Row i+1's 3 channel rows stream into the alternate LDS buffer
// (GLOBAL_LOAD_ASYNC_TO_LDS_B32, ASYNCcnt) while row i is gathered and
// stored; s_wait_asynccnt 3 retires only the oldest in-flight row.
// =====================================================================
__global__ void k4_crop(const float* __restrict__ img,
                        const float* __restrict__ fbx,
                        const int* __restrict__ tot,
                        float* __restrict__ out) {
  const int rc   = blockIdx.x;           // row chunk (16 rows)
  const int crop = blockIdx.y;           // b*K_ + kk
  const int b    = crop / K_;
  const int kk   = crop % K_;
  const int x    = threadIdx.x;          // output col
  const int r0   = rc * ROWS_PB;

  float* op0 = out + (size_t)crop * 3 * OUTSZ * OUTSZ + (size_t)r0 * OUTSZ + x;

  const bool valid = (kk == 0) || (kk == K_ - 1) || ((kk - 1) < tot[b]);
  if (!valid) {                          // uniform per block
#pragma unroll
    for (int i = 0; i < ROWS_PB; ++i) {
      float* op = op0 + (size_t)i * OUTSZ;
      op[0] = 0.0f;
      op[(size_t)OUTSZ * OUTSZ] = 0.0f;
      op[(size_t)2 * OUTSZ * OUTSZ] = 0.0f;
    }
    return;
  }

  const float* box = fbx + (size_t)(b * K_ + kk) * 4;
  const float inv = (float)OUTSZ / SCALE_F;
  int upper  = (int)(box[0] * inv); if (upper < 0) upper = 0;
  int left   = (int)(box[1] * inv); if (left  < 0) left  = 0;
  int bottom = (int)(box[2] * inv); if (bottom > 600) bottom = 600;  // ref: min(SCALE,.)
  int right  = (int)(box[3] * inv); if (right  > 600) right  = 600;
  int h = bottom - upper; if (h < 1) h = 1;
  int w = right  - left;  if (w < 1) w = 1;

  int col = left + (x * w) / OUTSZ;                   // row-invariant gather index
  if (col > OUTSZ - 1) col = OUTSZ - 1; if (col < 0) col = 0;

  __shared__ float s_row[2][3 * OUTSZ];
  const float* base = img + (size_t)b * 3 * OUTSZ * OUTSZ;

  // Issue async stage of one source row (3 channels) into LDS buffer `buf`.
  auto issue = [&](int buf, int r) {
    int rowsrc = upper + (r * h) / OUTSZ;
    if (rowsrc > OUTSZ - 1) rowsrc = OUTSZ - 1; if (rowsrc < 0) rowsrc = 0;
    unsigned lds0 = (unsigned)(size_t)(&s_row[buf][x]);  // LDS byte addr (low 32b)
    unsigned go0  = ((unsigned)rowsrc * OUTSZ + (unsigned)x) * 4u;
#pragma unroll
    for (int ch = 0; ch < 3; ++ch) {
      unsigned lds = lds0 + (unsigned)(ch * OUTSZ) * 4u;
      unsigned go  = go0 + (unsigned)(ch * OUTSZ * OUTSZ) * 4u;
      asm volatile("global_load_async_to_lds_b32 %0, %1, %2"
                   :: "v"(lds), "v"(go), "s"(base) : "memory");
    }
  };

  issue(0, r0);                                        // prime the pipeline
  for (int i = 0; i < ROWS_PB; ++i) {
    if (i + 1 < ROWS_PB) {
      issue((i + 1) & 1, r0 + i + 1);                  // overlap next-row fetch
      // 6 async ops in flight per wave; retire the oldest row's 3 (in-order)
      asm volatile("s_wait_asynccnt 0x3" ::: "memory");
    } else {
      asm volatile("s_wait_asynccnt 0x0" ::: "memory");
    }
    __syncthreads();                                   // all waves staged row i

    const float* sb = s_row[i & 1];
    float* op = op0 + (size_t)i * OUTSZ;
    op[0]                         = sb[col];
    op[(size_t)OUTSZ * OUTSZ]     = sb[OUTSZ + col];
    op[(size_t)2 * OUTSZ * OUTSZ] = sb[2 * OUTSZ + col];

    __syncthreads();                                   // protect buf reuse next iter
  }
}

// =====================================================================
extern "C" void kernel_launch(void* const* d_in, const int* in_sizes, int n_in,
                              void* d_out, int out_size, void* d_ws, size_t ws_size,
                              hipStream_t stream) {
  const float* boxes     = (const float*)d_in[0];
  const float* scores    = (const float*)d_in[1];
  const float* bbox_pred = (const float*)d_in[2];
  const float* images    = (const float*)d_in[3];
  float* out = (float*)d_out;

  float* pred = (float*)d_ws;
  int*   cnt  = (int*)(pred + PRED_ELEMS);
  float* kept = (float*)(cnt + CNT_ELEMS);
  float* fbx  = kept + KEPT_ELEMS;
  int*   tot  = (int*)(fbx + FBX_ELEMS);

  (void)in_sizes; (void)n_in; (void)out_size; (void)ws_size;

  k1_transform<<<(B_ * NBOX * NCLS + 255) / 256, 256, 0, stream>>>(boxes, bbox_pred, pred);
  k2_sort_nms<<<B_ * NC, 256, 0, stream>>>(scores, pred, kept, cnt);
  k3_assemble<<<B_, 128, 0, stream>>>(kept, cnt, fbx, tot, out);
  dim3 g4(OUTSZ / ROWS_PB, B_ * K_);
  k4_crop<<<g4, OUTSZ, 0, stream>>>(images, fbx, tot, out);
}